// SSGConv_3178275799599
// MI455X (gfx1250) — compile-verified
//
#include <hip/hip_runtime.h>
#include <hip/hip_bf16.h>

#define D_FEAT   64
#define K_HOPS   16
#define ALPHA_C  0.1f
// (1-ALPHA)/K
#define C1_C     (0.9f / 16.0f)

typedef __attribute__((ext_vector_type(2))) float v2f;
typedef __attribute__((ext_vector_type(8))) float v8f;

// ---------------------------------------------------------------------------
// CSR build: histogram
// ---------------------------------------------------------------------------
__global__ void count_kernel(const int* __restrict__ rows, int* __restrict__ cnt, int E) {
    for (int e = blockIdx.x * blockDim.x + threadIdx.x; e < E; e += gridDim.x * blockDim.x)
        atomicAdd(&cnt[rows[e]], 1);
}

// Single-block two-level exclusive scan of cnt[0..n) -> row_ptr[0..n]
__global__ void scan_kernel(const int* __restrict__ cnt, int* __restrict__ row_ptr, int n) {
    __shared__ int sums[1024];
    const int t = threadIdx.x;
    const int chunk = (n + 1023) >> 10;
    int b = t * chunk;
    int e = b + chunk; if (e > n) e = n;
    int s = 0;
    for (int i = b; i < e; ++i) s += cnt[i];
    sums[t] = s;
    __syncthreads();
    // Hillis-Steele inclusive scan (double-barrier safe)
    for (int off = 1; off < 1024; off <<= 1) {
        int v = 0;
        if (t >= off) v = sums[t - off];
        __syncthreads();
        if (t >= off) sums[t] += v;
        __syncthreads();
    }
    int excl = (t == 0) ? 0 : sums[t - 1];
    for (int i = b; i < e; ++i) { row_ptr[i] = excl; excl += cnt[i]; }
    if (t == 1023) row_ptr[n] = sums[1023];   // total nnz
}

__global__ void scatter_kernel(const int* __restrict__ rows, const int* __restrict__ row_ptr,
                               int* __restrict__ fill, int* __restrict__ eidx, int E) {
    for (int e = blockIdx.x * blockDim.x + threadIdx.x; e < E; e += gridDim.x * blockDim.x) {
        int r = rows[e];
        int p = row_ptr[r] + atomicAdd(&fill[r], 1);
        eidx[p] = e;
    }
}

// Restore deterministic (edge-index) order inside each row segment, then emit
// the packed (col, weight) edge stream. Avg segment len = E/N = 16.
__global__ void sort_emit_kernel(const int* __restrict__ row_ptr, int* __restrict__ eidx,
                                 const int* __restrict__ ecol, const float* __restrict__ ew,
                                 int2* __restrict__ epk, int n) {
    int r = blockIdx.x * blockDim.x + threadIdx.x;
    if (r >= n) return;
    int b = row_ptr[r], e = row_ptr[r + 1];
    if (e - b <= 512) {                       // guard against pathological rows
        for (int i = b + 1; i < e; ++i) {
            int key = eidx[i];
            int j = i - 1;
            while (j >= b && eidx[j] > key) { eidx[j + 1] = eidx[j]; --j; }
            eidx[j + 1] = key;
        }
    }
    for (int i = b; i < e; ++i) {
        int ed = eidx[i];
        int2 p;
        p.x = ecol[ed];
        p.y = __float_as_int(ew[ed]);
        epk[i] = p;
    }
}

// ---------------------------------------------------------------------------
// SpMM: one wave32 per output row; lane holds 2 features (float2).
// Edge stream is wave-uniform: row is pushed through readfirstlane so the
// backend can keep row_ptr / packed-edge loads on the scalar path and
// dedicate the vector memory pipe to the 256B L2-resident row gathers.
// Also accumulates S += h_out (running sum over hops). Zero atomics.
// ---------------------------------------------------------------------------
__global__ void spmm_kernel(const float2* __restrict__ hin, float2* __restrict__ hout,
                            float2* __restrict__ S,
                            const int* __restrict__ row_ptr,
                            const int2* __restrict__ epk,
                            int n_nodes) {
    int row  = (blockIdx.x * blockDim.x + threadIdx.x) >> 5;  // wave-uniform
    int lane = threadIdx.x & 31;
    if (row >= n_nodes) return;
    row = __builtin_amdgcn_readfirstlane(row);                // provably uniform (SGPR)
    int beg = row_ptr[row], end = row_ptr[row + 1];
    float ax = 0.0f, ay = 0.0f;
    for (int j = beg; j < end; ++j) {
        int2  p = epk[j];                 // uniform 8B load: (col, weight)
        int   c = p.x;
        float w = __int_as_float(p.y);
        float2 v = hin[c * 32 + lane];    // coalesced 256B row read (L2-resident)
        ax = fmaf(w, v.x, ax);
        ay = fmaf(w, v.y, ay);
    }
    int o = row * 32 + lane;
    float2 acc; acc.x = ax; acc.y = ay;
    hout[o] = acc;
    float2 s = S[o];
    s.x += ax; s.y += ay;
    S[o] = s;
}

// ---------------------------------------------------------------------------
// Epilogue: out = C1*S + ALPHA*x, computed per 16x16 tile with
// V_WMMA_F32_16X16X4_F32:  D = sum_j (C1*I[:,4j:4j+4]) x S[4j:4j+4,:] + (ALPHA*x)
// Exact: identity columns pick rows; zero products add exactly.
// Grid is exact (N%16==0), wid guard is wave-uniform -> EXEC all ones at WMMA.
// ---------------------------------------------------------------------------
__global__ void epilogue_kernel(const float* __restrict__ S, const float* __restrict__ x,
                                float* __restrict__ out, int n_tiles) {
    int wid  = (blockIdx.x * blockDim.x + threadIdx.x) >> 5;  // wave-uniform
    int lane = threadIdx.x & 31;
    if (wid >= n_tiles) return;                               // wave-uniform branch
    int tn = wid & 3;            // feature-block 0..3
    int tm = wid >> 2;           // node-block
    int m0 = tm * 16, n0 = tn * 16;
    int half = lane >> 4;        // 0: lanes 0-15, 1: lanes 16-31
    int lm   = lane & 15;

    // C/D layout: vgpr r, lane -> (M = r + 8*half, N = lm)
    v8f c;
#pragma unroll
    for (int r = 0; r < 8; ++r)
        c[r] = ALPHA_C * x[(m0 + r + 8 * half) * D_FEAT + n0 + lm];

#pragma unroll
    for (int j = 0; j < 4; ++j) {
        // A layout (16x4 f32): elem e, lane -> (M = lm, K = e + 2*half)
        int k0 = 2 * half;       // K of elem0
        int k1 = 1 + 2 * half;   // K of elem1
        v2f a, b;
        a[0] = (lm == 4 * j + k0) ? C1_C : 0.0f;
        a[1] = (lm == 4 * j + k1) ? C1_C : 0.0f;
        // B layout (4x16 f32), mirrored: elem e, lane -> (K = e + 2*half, N = lm)
        b[0] = S[(m0 + 4 * j + k0) * D_FEAT + n0 + lm];
        b[1] = S[(m0 + 4 * j + k1) * D_FEAT + n0 + lm];
        c = __builtin_amdgcn_wmma_f32_16x16x4_f32(false, a, false, b, (short)0, c,
                                                  false, false);
    }
#pragma unroll
    for (int r = 0; r < 8; ++r)
        out[(m0 + r + 8 * half) * D_FEAT + n0 + lm] = c[r];
}

// ---------------------------------------------------------------------------
extern "C" void kernel_launch(void* const* d_in, const int* in_sizes, int n_in,
                              void* d_out, int out_size, void* d_ws, size_t ws_size,
                              hipStream_t stream) {
    const float* x   = (const float*)d_in[0];
    const int*   er  = (const int*)d_in[1];
    const int*   ec  = (const int*)d_in[2];
    const float* ew  = (const float*)d_in[3];
    float*       out = (float*)d_out;

    const int N = in_sizes[0] / D_FEAT;   // 100000
    const int E = in_sizes[1];            // 1600000

    // ---- workspace layout (256B aligned) ----
    char* p = (char*)d_ws;
    auto take = [&](size_t bytes) -> char* {
        char* q = p;
        p += (bytes + 255) & ~size_t(255);
        return q;
    };
    float* hA      = (float*)take((size_t)N * D_FEAT * sizeof(float));
    float* hB      = (float*)take((size_t)N * D_FEAT * sizeof(float));
    float* Ssum    = (float*)take((size_t)N * D_FEAT * sizeof(float));
    int*   row_ptr = (int*)  take((size_t)(N + 1) * sizeof(int));
    int*   cnt     = (int*)  take((size_t)N * sizeof(int));        // reused as fill
    int*   eidx    = (int*)  take((size_t)E * sizeof(int));
    int2*  epk     = (int2*) take((size_t)E * sizeof(int2));       // packed (col, w)

    // ---- CSR build ----
    hipMemsetAsync(cnt, 0, (size_t)N * sizeof(int), stream);
    count_kernel<<<2048, 256, 0, stream>>>(er, cnt, E);
    scan_kernel<<<1, 1024, 0, stream>>>(cnt, row_ptr, N);
    hipMemsetAsync(cnt, 0, (size_t)N * sizeof(int), stream);       // now "fill"
    scatter_kernel<<<2048, 256, 0, stream>>>(er, row_ptr, cnt, eidx, E);
    sort_emit_kernel<<<(N + 255) / 256, 256, 0, stream>>>(row_ptr, eidx, ec, ew,
                                                          epk, N);

    // ---- K hops, accumulating Ssum ----
    hipMemsetAsync(Ssum, 0, (size_t)N * D_FEAT * sizeof(float), stream);
    const int spmm_blocks = (N * 32 + 255) / 256;   // wave per row, 8 waves/block
    for (int it = 0; it < K_HOPS; ++it) {
        const float* hin  = (it == 0) ? x : ((it & 1) ? hA : hB);
        float*       hout = (it & 1) ? hB : hA;
        spmm_kernel<<<spmm_blocks, 256, 0, stream>>>(
            (const float2*)hin, (float2*)hout, (float2*)Ssum,
            row_ptr, epk, N);
    }

    // ---- WMMA epilogue ----
    const int n_tiles = (N / 16) * (D_FEAT / 16);   // 25000, exact
    const int epi_blocks = (n_tiles * 32 + 255) / 256;
    epilogue_kernel<<<epi_blocks, 256, 0, stream>>>(Ssum, x, out, n_tiles);
}